// CapsuleLayer_11596411699768
// MI455X (gfx1250) — compile-verified
//
#include <hip/hip_runtime.h>
#include <math.h>

// Problem constants (from reference)
#define BB   256          // batch
#define IU   8            // in_units (i)
#define JC   1152         // in_channels (j)
#define NN   10           // num_units (n)
#define UU   16           // unit_size (u)
#define NU   (NN * UU)    // 160
#define KTOT (IU * JC)    // 9216  (k = i*JC + j  -> A operand == x, contiguous)
#define KSPLIT 48
#define KCHUNK (KTOT / KSPLIT)   // 192
#define NTPW1 5                  // N-tiles per wave, GEMM1 (160/16 = 10 = 2 groups * 5)
#define NTPW2 4                  // N-tiles per wave, GEMM2 (576 tiles = 144 groups * 4)

typedef __attribute__((ext_vector_type(2))) float v2f;
typedef __attribute__((ext_vector_type(8))) float v8f;

// ---------------------------------------------------------------------------
__global__ void init_zero_kernel(float* __restrict__ p, int n) {
    int e = blockIdx.x * blockDim.x + threadIdx.x;
    if (e < n) p[e] = 0.0f;
}

// One-time LDS-tiled transpose: XT[k][b] = X[b][k].  grid (KTOT/32, BB/32), block (32,8).
__global__ __launch_bounds__(256) void transpose_x_kernel(const float* __restrict__ X,
                                                          float* __restrict__ XT) {
    __shared__ float tile[32][33];
    const int k0 = blockIdx.x * 32;
    const int b0 = blockIdx.y * 32;
#pragma unroll
    for (int yy = threadIdx.y; yy < 32; yy += 8)
        tile[yy][threadIdx.x] = X[(size_t)(b0 + yy) * KTOT + k0 + threadIdx.x];
    __syncthreads();
#pragma unroll
    for (int yy = threadIdx.y; yy < 32; yy += 8)
        XT[(size_t)(k0 + yy) * BB + b0 + threadIdx.x] = tile[threadIdx.x][yy];
}

// softmax over j (axis 0) for each n: c[j,n] = softmax_j(blog[j,n]).  grid = NN blocks.
__global__ __launch_bounds__(256) void softmax_j_kernel(const float* __restrict__ blog,
                                                        float* __restrict__ c) {
    __shared__ float red[256];
    const int n = blockIdx.x;
    const int tid = threadIdx.x;
    float m = -3.4e38f;
    for (int j = tid; j < JC; j += 256) m = fmaxf(m, blog[j * NN + n]);
    red[tid] = m; __syncthreads();
    for (int s = 128; s > 0; s >>= 1) { if (tid < s) red[tid] = fmaxf(red[tid], red[tid + s]); __syncthreads(); }
    const float mx = red[0]; __syncthreads();
    float sum = 0.0f;
    for (int j = tid; j < JC; j += 256) sum += expf(blog[j * NN + n] - mx);
    red[tid] = sum; __syncthreads();
    for (int s = 128; s > 0; s >>= 1) { if (tid < s) red[tid] += red[tid + s]; __syncthreads(); }
    const float inv = 1.0f / red[0];
    for (int j = tid; j < JC; j += 256) c[j * NN + n] = expf(blog[j * NN + n] - mx) * inv;
}

// WcT[nu=(n*UU+u)][k=(i*JC+j)] = c[j,n] * W[j,n,u,i].  (transposed for b64 B-frag loads)
__global__ __launch_bounds__(256) void scale_w_kernel(const float* __restrict__ c,
                                                      const float* __restrict__ W,
                                                      float* __restrict__ WcT) {
    const int e  = blockIdx.x * blockDim.x + threadIdx.x;  // < NU*KTOT = 1474560
    const int nu = e / KTOT;
    const int k  = e - nu * KTOT;
    const int i  = k / JC;
    const int j  = k - i * JC;
    const int n  = nu / UU;
    const int u  = nu - n * UU;
    WcT[e] = c[j * NN + n] * W[(((size_t)(j * NN + n)) * UU + u) * IU + i];
}

// GEMM1 (K-split partials): s_part[split][b][nu] = sum_{k in chunk} X[b,k]*Wc[k,nu]
// wave = (split, tM, ng); each wave: 1 A-frag vs NTPW1 B-frags per k-step.
__global__ __launch_bounds__(256) void gemm_s_partial_kernel(const float* __restrict__ X,
                                                             const float* __restrict__ WcT,
                                                             float* __restrict__ s_part) {
    const int lane  = threadIdx.x & 31;
    const int gwave = blockIdx.x * (blockDim.x >> 5) + (threadIdx.x >> 5);
    const int split = gwave >> 5;            // / 32  (16 tM * 2 ng)
    const int rem   = gwave & 31;
    const int tM    = rem >> 1;
    const int ng    = rem & 1;
    const int l15   = lane & 15;
    const int khalf = (lane >> 4) << 1;      // 0 or 2 (ISA f32 A/B fragment layout)
    const int mrow  = tM * 16 + l15;
    const int kbase = split * KCHUNK + khalf;

    const float* Xrow = X + (size_t)mrow * KTOT + kbase;
    const float* Bp[NTPW1];
#pragma unroll
    for (int t = 0; t < NTPW1; ++t)
        Bp[t] = WcT + (size_t)(ng * (NTPW1 * 16) + t * 16 + l15) * KTOT + kbase;

    const v8f z = {0.f, 0.f, 0.f, 0.f, 0.f, 0.f, 0.f, 0.f};
    v8f acc[NTPW1];
#pragma unroll
    for (int t = 0; t < NTPW1; ++t) acc[t] = z;

#pragma unroll 4
    for (int kk = 0; kk < KCHUNK; kk += 4) {
        const v2f a = *(const v2f*)(Xrow + kk);            // A[m,k0], A[m,k0+1]
#pragma unroll
        for (int t = 0; t < NTPW1; ++t) {
            const v2f b = *(const v2f*)(Bp[t] + kk);       // B[k0,n], B[k0+1,n]
            acc[t] = __builtin_amdgcn_wmma_f32_16x16x4_f32(false, a, false, b,
                                                           (short)0, acc[t], false, false);
        }
    }
    const int mbase = tM * 16 + ((lane >> 4) << 3);        // C/D: vgpr r -> row r / r+8
#pragma unroll
    for (int t = 0; t < NTPW1; ++t) {
        const int ncol = ng * (NTPW1 * 16) + t * 16 + l15;
        float* dst = s_part + ((size_t)split * BB + mbase) * NU + ncol;
#pragma unroll
        for (int r = 0; r < 8; ++r) dst[(size_t)r * NU] = acc[t][r];
    }
}

// Deterministic fixed-order reduction of the K-split partials. 160 blocks * 256 = 40960.
__global__ __launch_bounds__(256) void reduce_s_kernel(const float* __restrict__ s_part,
                                                       float* __restrict__ s) {
    const int e = blockIdx.x * blockDim.x + threadIdx.x;   // < BB*NU
    float acc = 0.0f;
#pragma unroll 8
    for (int sp = 0; sp < KSPLIT; ++sp) acc += s_part[(size_t)sp * BB * NU + e];
    s[e] = acc;
}

// Squash: msq[b,u] = sum_n s[b,n,u]^2 ; v = s*sqrt(msq)/(1+msq).
// Writes vT[nu][b] (GEMM2 A operand); optionally also out[b][nu] on the last iteration.
__global__ __launch_bounds__(256) void squash_kernel(const float* __restrict__ s,
                                                     float* __restrict__ vT,
                                                     float* __restrict__ out,
                                                     int write_out) {
    const int e = blockIdx.x * blockDim.x + threadIdx.x;   // < BB*UU = 4096
    const int b = e >> 4;
    const int u = e & 15;
    float sv[NN];
    float msq = 0.0f;
#pragma unroll
    for (int n = 0; n < NN; ++n) {
        float t = s[(size_t)b * NU + n * UU + u];
        sv[n] = t;
        msq += t * t;
    }
    const float scale = sqrtf(msq) / (1.0f + msq);
#pragma unroll
    for (int n = 0; n < NN; ++n) {
        const float val = sv[n] * scale;
        vT[(size_t)(n * UU + u) * BB + b] = val;
        if (write_out) out[(size_t)b * NU + n * UU + u] = val;
    }
}

// GEMM2: T[nu, k] = sum_b v[b,nu] * X[b,k].  A = vT[nu][b], B = XT[k][b]: all b64 frag loads.
// wave = (tM in 0..9, ng in 0..143); NTPW2 = 4 N-tiles per wave; K = 256 (full).
__global__ __launch_bounds__(256) void gemm_T_kernel(const float* __restrict__ vT,
                                                     const float* __restrict__ XT,
                                                     float* __restrict__ T) {
    const int lane  = threadIdx.x & 31;
    const int gwave = blockIdx.x * (blockDim.x >> 5) + (threadIdx.x >> 5);
    const int tM    = gwave / 144;
    const int ng    = gwave - tM * 144;
    const int l15   = lane & 15;
    const int khalf = (lane >> 4) << 1;
    const int m     = tM * 16 + l15;                       // nu row

    const float* Ap = vT + (size_t)m * BB + khalf;         // A[m,b] pairs along b
    const float* Bp[NTPW2];
#pragma unroll
    for (int t = 0; t < NTPW2; ++t)
        Bp[t] = XT + (size_t)((ng * NTPW2 + t) * 16 + l15) * BB + khalf;  // B[b,k] pairs along b

    const v8f z = {0.f, 0.f, 0.f, 0.f, 0.f, 0.f, 0.f, 0.f};
    v8f acc[NTPW2];
#pragma unroll
    for (int t = 0; t < NTPW2; ++t) acc[t] = z;

#pragma unroll 4
    for (int b0 = 0; b0 < BB; b0 += 4) {
        const v2f a = *(const v2f*)(Ap + b0);
#pragma unroll
        for (int t = 0; t < NTPW2; ++t) {
            const v2f b = *(const v2f*)(Bp[t] + b0);
            acc[t] = __builtin_amdgcn_wmma_f32_16x16x4_f32(false, a, false, b,
                                                           (short)0, acc[t], false, false);
        }
    }
    const int mbase = tM * 16 + ((lane >> 4) << 3);
#pragma unroll
    for (int t = 0; t < NTPW2; ++t) {
        const int col = (ng * NTPW2 + t) * 16 + l15;
        float* dst = T + (size_t)mbase * KTOT + col;
#pragma unroll
        for (int r = 0; r < 8; ++r) dst[(size_t)r * KTOT] = acc[t][r];
    }
}

// blog[j,n] += (1/B) * sum_{u,i} W[j,n,u,i] * T[(n,u), i*JC + j].  45 blocks * 256 = 11520.
__global__ __launch_bounds__(256) void bupdate_kernel(const float* __restrict__ W,
                                                      const float* __restrict__ T,
                                                      float* __restrict__ blog) {
    const int e = blockIdx.x * blockDim.x + threadIdx.x;   // e = j*NN + n
    const int j = e / NN;
    const int n = e - j * NN;
    const float* Wjn = W + (size_t)e * (UU * IU);
    float acc = 0.0f;
#pragma unroll
    for (int u = 0; u < UU; ++u) {
        const float* Trow = T + (size_t)(n * UU + u) * KTOT + j;
#pragma unroll
        for (int i = 0; i < IU; ++i)
            acc += Wjn[u * IU + i] * Trow[(size_t)i * JC];
    }
    blog[e] += acc * (1.0f / (float)BB);
}

// ---------------------------------------------------------------------------
extern "C" void kernel_launch(void* const* d_in, const int* in_sizes, int n_in,
                              void* d_out, int out_size, void* d_ws, size_t ws_size,
                              hipStream_t stream) {
    (void)in_sizes; (void)n_in; (void)out_size; (void)ws_size;
    const float* X = (const float*)d_in[0];   // [256][8][1152] == [B][KTOT] with k=i*JC+j
    const float* W = (const float*)d_in[1];   // [1152][10][16][8]
    float* out = (float*)d_out;               // [256][10][16][1] = [B][NU]
    float* ws = (float*)d_ws;

    float* blog   = ws;                                  // JC*NN          = 11520
    float* c      = blog   + (size_t)JC * NN;            // JC*NN          = 11520
    float* WcT    = c      + (size_t)JC * NN;            // NU*KTOT        = 1474560
    float* s      = WcT    + (size_t)NU * KTOT;          // BB*NU          = 40960
    float* vT     = s      + (size_t)BB * NU;            // NU*BB          = 40960
    float* T      = vT     + (size_t)NU * BB;            // NU*KTOT        = 1474560
    float* XT     = T      + (size_t)NU * KTOT;          // KTOT*BB        = 2359296
    float* s_part = XT     + (size_t)KTOT * BB;          // KSPLIT*BB*NU   = 1966080
    // total ~7.4M floats (~29.5 MB)

    init_zero_kernel<<<45, 256, 0, stream>>>(blog, JC * NN);
    transpose_x_kernel<<<dim3(KTOT / 32, BB / 32), dim3(32, 8), 0, stream>>>(X, XT);

    for (int it = 0; it < 3; ++it) {
        softmax_j_kernel<<<NN, 256, 0, stream>>>(blog, c);
        scale_w_kernel<<<(NU * KTOT) / 256, 256, 0, stream>>>(c, W, WcT);
        gemm_s_partial_kernel<<<(16 * 2 * KSPLIT) / 8, 256, 0, stream>>>(X, WcT, s_part);
        reduce_s_kernel<<<(BB * NU) / 256, 256, 0, stream>>>(s_part, s);
        squash_kernel<<<(BB * UU) / 256, 256, 0, stream>>>(s, vT, out, (it == 2) ? 1 : 0);
        if (it < 2) {
            gemm_T_kernel<<<(10 * 144) / 8, 256, 0, stream>>>(vT, XT, T);
            bupdate_kernel<<<(JC * NN) / 256, 256, 0, stream>>>(W, T, blog);
        }
    }
}